// TorchGaussianHMM_33174327394898
// MI455X (gfx1250) — compile-verified
//
#include <hip/hip_runtime.h>
#include <math.h>

#define TLEN 131072
#define NST 32
#define FD 16
#define CHUNK 128
#define NCHUNK (TLEN / CHUNK)   // 1024

typedef __attribute__((ext_vector_type(2))) float v2f;
typedef __attribute__((ext_vector_type(8))) float v8f;

__device__ __forceinline__ float warp_max32(float v) {
#pragma unroll
    for (int off = 16; off >= 1; off >>= 1)
        v = fmaxf(v, __shfl_xor(v, off, 32));
    return v;
}
__device__ __forceinline__ float warp_sum32(float v) {
#pragma unroll
    for (int off = 16; off >= 1; off >>= 1)
        v += __shfl_xor(v, off, 32);
    return v;
}

// ---------------------------------------------------------------------------
// Kernel 1: parameter prep. 32 threads, one HMM component per thread.
// Builds cov = tril(L)L^T + jitter*I, Cholesky, P = chol^{-1},
// b = P*mu, const_n = -0.5*(F*log2pi + logdet). Also softmax tables.
// ---------------------------------------------------------------------------
__global__ void hmm_prep(const float* __restrict__ uT, const float* __restrict__ uS,
                         const float* __restrict__ means, const float* __restrict__ ucc,
                         float* __restrict__ wPT, float* __restrict__ wB,
                         float* __restrict__ wC, float* __restrict__ wLogA,
                         float* __restrict__ wAlin, float* __restrict__ wLogPi) {
    int n = threadIdx.x;
    if (n >= NST) return;
    float ch[FD][FD];
    float p[FD][FD];
    const float* Ln = ucc + (size_t)n * FD * FD;
    // cov lower triangle (cov[i][j] = sum_{k<=min(i,j)} L[i][k]L[j][k])
    for (int i = 0; i < FD; ++i)
        for (int j = 0; j <= i; ++j) {
            float s = 0.f;
            for (int k = 0; k <= j; ++k) s += Ln[i * FD + k] * Ln[j * FD + k];
            if (i == j) s += 1e-4f;
            ch[i][j] = s;
        }
    // in-place Cholesky (lower)
    float logdet = 0.f;
    for (int j = 0; j < FD; ++j) {
        float d = ch[j][j];
        for (int k = 0; k < j; ++k) d -= ch[j][k] * ch[j][k];
        d = sqrtf(d);
        ch[j][j] = d;
        logdet += 2.f * logf(d);
        float inv = 1.f / d;
        for (int i = j + 1; i < FD; ++i) {
            float t = ch[i][j];
            for (int k = 0; k < j; ++k) t -= ch[i][k] * ch[j][k];
            ch[i][j] = t * inv;
        }
    }
    // invert lower triangular -> p
    for (int c = 0; c < FD; ++c) {
        for (int r = 0; r < FD; ++r) p[r][c] = 0.f;
        p[c][c] = 1.f / ch[c][c];
        for (int r = c + 1; r < FD; ++r) {
            float t = 0.f;
            for (int k = c; k < r; ++k) t += ch[r][k] * p[k][c];
            p[r][c] = -t / ch[r][r];
        }
    }
    // b = p * mu
    const float* mu = means + n * FD;
    for (int k = 0; k < FD; ++k) {
        float t = 0.f;
        for (int f = 0; f <= k; ++f) t += p[k][f] * mu[f];
        wB[n * FD + k] = t;
    }
    // PT[n][f][k] = p[k][f] (zero above diagonal so K=16 WMMA is exact)
    for (int f = 0; f < FD; ++f)
        for (int k = 0; k < FD; ++k)
            wPT[(size_t)n * FD * FD + f * FD + k] = (k >= f) ? p[k][f] : 0.f;
    wC[n] = -0.5f * (FD * 1.8378770664093453f + logdet);

    // row softmax of transition matrix; Alin = softmax + eps, logA = log(Alin)
    const float* row = uT + n * NST;
    float mx = row[0];
    for (int j = 1; j < NST; ++j) mx = fmaxf(mx, row[j]);
    float sum = 0.f;
    for (int j = 0; j < NST; ++j) sum += expf(row[j] - mx);
    float invs = 1.f / sum;
    for (int j = 0; j < NST; ++j) {
        float a = expf(row[j] - mx) * invs + 1e-8f;
        wAlin[n * NST + j] = a;
        wLogA[n * NST + j] = logf(a);
    }
    if (n == 0) {
        float m2 = uS[0];
        for (int j = 1; j < NST; ++j) m2 = fmaxf(m2, uS[j]);
        float s2 = 0.f;
        for (int j = 0; j < NST; ++j) s2 += expf(uS[j] - m2);
        float i2 = 1.f / s2;
        for (int j = 0; j < NST; ++j)
            wLogPi[j] = logf(expf(uS[j] - m2) * i2 + 1e-8f);
    }
}

// ---------------------------------------------------------------------------
// Kernel 2: emission log-probs via V_WMMA_F32_16X16X4_F32.
// One wave per 16-row tile of X. S = X_tile @ P_n^T, maha = ||S - b_n||^2 rows.
// Component loop unrolled x2 so two independent WMMA chains interleave.
// ---------------------------------------------------------------------------
__global__ void __launch_bounds__(256) hmm_emission(
    const float* __restrict__ X, const float* __restrict__ wPT,
    const float* __restrict__ wB, const float* __restrict__ wC,
    float* __restrict__ logE) {
    __shared__ float sPT[NST * FD * FD];   // 32 KB
    __shared__ float sB[NST * FD];
    __shared__ float sC[NST];
    for (int i = threadIdx.x; i < NST * FD * FD; i += 256) sPT[i] = wPT[i];
    for (int i = threadIdx.x; i < NST * FD; i += 256) sB[i] = wB[i];
    if (threadIdx.x < NST) sC[threadIdx.x] = wC[threadIdx.x];
    __syncthreads();

    int gw = (blockIdx.x * 256 + threadIdx.x) >> 5;   // tile id, 0..8191
    int lane = threadIdx.x & 31;
    int col = lane & 15;
    int kh = (lane >> 4) << 1;                        // K-half offset: 0 or 2
    long t0 = (long)gw * 16;

    // A operand (16x4 per chunk): lanes 0-15 & 16-31 both hold rows M=0..15,
    // VGPR0 = K (lo half) / K+2 (hi half), VGPR1 = K+1 / K+3.
    const float* Xr = X + (t0 + col) * FD;
    v2f a[4];
#pragma unroll
    for (int q = 0; q < 4; ++q) {
        a[q].x = Xr[4 * q + kh];
        a[q].y = Xr[4 * q + 1 + kh];
    }

#pragma unroll 2
    for (int n = 0; n < NST; ++n) {
        const float* pn = sPT + n * FD * FD;
        v8f acc = {0.f, 0.f, 0.f, 0.f, 0.f, 0.f, 0.f, 0.f};
#pragma unroll
        for (int q = 0; q < 4; ++q) {
            v2f b;   // B operand (4x16): mirrored K-half layout
            b.x = pn[(4 * q + kh) * FD + col];
            b.y = pn[(4 * q + 1 + kh) * FD + col];
            acc = __builtin_amdgcn_wmma_f32_16x16x4_f32(
                false, a[q], false, b, (short)0, acc, false, false);
        }
        float bs = sB[n * FD + col];
        float cst = sC[n];
#pragma unroll
        for (int v = 0; v < 8; ++v) {
            float d = acc[v] - bs;
            float sq = d * d;
            sq += __shfl_xor(sq, 1, 32);
            sq += __shfl_xor(sq, 2, 32);
            sq += __shfl_xor(sq, 4, 32);
            sq += __shfl_xor(sq, 8, 32);
            float le = cst - 0.5f * sq;
            int row = v + ((lane >> 4) << 3);
            if (col == v) logE[(t0 + row) * NST + n] = le;
        }
    }
}

// ---------------------------------------------------------------------------
// Kernel 3: phase 1 of the scan. One wave per chunk of 128 timesteps.
// Builds linear chunk matrix M_c = prod_t (A * diag(e_t)) with per-step
// max-rescaling; stores log(M_c) and accumulated log-scale.
// Lane l holds row l of M in registers. logE rows are prefetched ahead
// (global_prefetch_b8) to hide load latency on the serial critical path.
// ---------------------------------------------------------------------------
__global__ void __launch_bounds__(256) hmm_scan_chunks(
    const float* __restrict__ logE, const float* __restrict__ Alin,
    float* __restrict__ chunkLogM, float* __restrict__ chunkS) {
    __shared__ float sA[NST * NST];
    for (int i = threadIdx.x; i < NST * NST; i += 256) sA[i] = Alin[i];
    __syncthreads();
    int c = (blockIdx.x * 256 + threadIdx.x) >> 5;
    int lane = threadIdx.x & 31;
    float m[NST];
#pragma unroll
    for (int j = 0; j < NST; ++j) m[j] = (lane == j) ? 1.f : 0.f;
    float s = 0.f;
    int tstart = c * CHUNK;
    for (int t = tstart; t < tstart + CHUNK; ++t) {
        // speculative prefetch ~16 steps ahead (stays inside workspace)
        __builtin_prefetch(&logE[(long)(t + 16) * NST + lane], 0, 0);
        float le = logE[(long)t * NST + lane];
        float mt = warp_max32(le);
        float e = expf(le - mt);
        if (t == 0) {           // chunk 0: alpha0 = pi .* e_0 (no transition)
#pragma unroll
            for (int j = 0; j < NST; ++j) m[j] *= __shfl(e, j, 32);
            s += mt;            // max entry of diag(e) is 1, no renorm needed
            continue;
        }
        float nw[NST];
#pragma unroll
        for (int j = 0; j < NST; ++j) {
            float acc = 0.f;
#pragma unroll
            for (int k = 0; k < NST; ++k)
                acc = fmaf(m[k], sA[k * NST + j], acc);
            nw[j] = acc * __shfl(e, j, 32);
        }
        float g = nw[0];
#pragma unroll
        for (int j = 1; j < NST; ++j) g = fmaxf(g, nw[j]);
        g = warp_max32(g);      // g >= 1e-8 guaranteed by the +eps floor on A
        float inv = 1.f / g;
#pragma unroll
        for (int j = 0; j < NST; ++j) m[j] = nw[j] * inv;
        s += mt + logf(g);
    }
#pragma unroll
    for (int j = 0; j < NST; ++j)
        chunkLogM[(long)c * (NST * NST) + lane * NST + j] = logf(m[j]);
    if (lane == 0) chunkS[c] = s;
}

// ---------------------------------------------------------------------------
// Kernel 4: phase 2. One wave sequentially composes the 1024 chunk matrices
// in log space, recording each chunk's incoming alpha vector.
// Next chunk's 4KB matrix is prefetched (32 lanes x 128B = full matrix).
// ---------------------------------------------------------------------------
__global__ void hmm_combine(const float* __restrict__ wLogPi,
                            const float* __restrict__ chunkLogM,
                            const float* __restrict__ chunkS,
                            float* __restrict__ chain) {
    int lane = threadIdx.x;
    float v = wLogPi[lane];
    for (int c = 0; c < NCHUNK; ++c) {
        const float* M = chunkLogM + (long)c * (NST * NST);
        // prefetch next chunk's matrix: lane-staggered, one 128B line per lane
        __builtin_prefetch(M + NST * NST + lane * NST, 0, 0);
        chain[c * NST + lane] = v;
        float x[NST];
        float mx = -__builtin_inff();
#pragma unroll
        for (int i = 0; i < NST; ++i) {
            float xi = __shfl(v, i, 32) + M[i * NST + lane];
            x[i] = xi;
            mx = fmaxf(mx, xi);
        }
        float nv;
        if (mx == -__builtin_inff()) {
            nv = -__builtin_inff();
        } else {
            float sum = 0.f;
#pragma unroll
            for (int i = 0; i < NST; ++i) sum += expf(x[i] - mx);
            nv = mx + logf(sum);
        }
        v = nv + chunkS[c];
    }
}

// ---------------------------------------------------------------------------
// Kernel 5: phase 3. One wave per chunk replays the exact log-space
// recursion (same math as the reference scan) and writes alpha.
// ---------------------------------------------------------------------------
__global__ void __launch_bounds__(256) hmm_alpha(
    const float* __restrict__ logE, const float* __restrict__ wLogA,
    const float* __restrict__ chain, float* __restrict__ alpha) {
    __shared__ float sA[NST * NST];
    for (int i = threadIdx.x; i < NST * NST; i += 256) sA[i] = wLogA[i];
    __syncthreads();
    int c = (blockIdx.x * 256 + threadIdx.x) >> 5;
    int lane = threadIdx.x & 31;
    float a = chain[c * NST + lane];
    int tstart = c * CHUNK;
    for (int t = tstart; t < tstart + CHUNK; ++t) {
        __builtin_prefetch(&logE[(long)(t + 16) * NST + lane], 0, 0);
        float le = logE[(long)t * NST + lane];
        if (t == 0) {
            a += le;
        } else {
            float x[NST];
            float mx = -__builtin_inff();
#pragma unroll
            for (int i = 0; i < NST; ++i) {
                float xi = __shfl(a, i, 32) + sA[i * NST + lane];
                x[i] = xi;
                mx = fmaxf(mx, xi);
            }
            if (mx == -__builtin_inff()) {
                a = -__builtin_inff();
            } else {
                float sum = 0.f;
#pragma unroll
                for (int i = 0; i < NST; ++i) sum += expf(x[i] - mx);
                a = mx + logf(sum) + le;
            }
        }
        alpha[(long)t * NST + lane] = a;
    }
}

// ---------------------------------------------------------------------------
// Kernel 6: final logsumexp over alpha[T-1] -> log_prob.
// ---------------------------------------------------------------------------
__global__ void hmm_final(const float* __restrict__ alpha, float* __restrict__ out0) {
    int lane = threadIdx.x;
    float a = alpha[(long)(TLEN - 1) * NST + lane];
    float mx = warp_max32(a);
    float s = warp_sum32(expf(a - mx));
    if (lane == 0) out0[0] = mx + logf(s);
}

extern "C" void kernel_launch(void* const* d_in, const int* in_sizes, int n_in,
                              void* d_out, int out_size, void* d_ws, size_t ws_size,
                              hipStream_t stream) {
    (void)in_sizes; (void)n_in; (void)out_size; (void)ws_size;
    const float* X     = (const float*)d_in[0];   // (T,F)
    const float* uT    = (const float*)d_in[1];   // (N,N)
    const float* uS    = (const float*)d_in[2];   // (N,)
    const float* means = (const float*)d_in[3];   // (N,F)
    const float* ucc   = (const float*)d_in[4];   // (N,F,F)
    float* out = (float*)d_out;                   // [log_prob, alpha(T,N)]
    float* ws = (float*)d_ws;

    size_t off = 0;
    float* wLogE   = ws + off; off += (size_t)TLEN * NST;         // 4,194,304
    float* wPT     = ws + off; off += (size_t)NST * FD * FD;      // 8,192
    float* wB      = ws + off; off += (size_t)NST * FD;           // 512
    float* wC      = ws + off; off += NST;                        // 32
    float* wLogA   = ws + off; off += NST * NST;                  // 1,024
    float* wAlin   = ws + off; off += NST * NST;                  // 1,024
    float* wLogPi  = ws + off; off += NST;                        // 32
    float* wChunkM = ws + off; off += (size_t)NCHUNK * NST * NST; // 1,048,576
    float* wChunkS = ws + off; off += NCHUNK;                     // 1,024
    float* wChain  = ws + off; off += (size_t)NCHUNK * NST;       // 32,768

    hmm_prep<<<1, 32, 0, stream>>>(uT, uS, means, ucc, wPT, wB, wC, wLogA, wAlin, wLogPi);
    hmm_emission<<<(TLEN / 16) * 32 / 256, 256, 0, stream>>>(X, wPT, wB, wC, wLogE);
    hmm_scan_chunks<<<NCHUNK * 32 / 256, 256, 0, stream>>>(wLogE, wAlin, wChunkM, wChunkS);
    hmm_combine<<<1, 32, 0, stream>>>(wLogPi, wChunkM, wChunkS, wChain);
    hmm_alpha<<<NCHUNK * 32 / 256, 256, 0, stream>>>(wLogE, wLogA, wChain, out + 1);
    hmm_final<<<1, 32, 0, stream>>>(out + 1, out);
}